// TestNet55_23055384445042
// MI455X (gfx1250) — compile-verified
//
#include <hip/hip_runtime.h>
#include <math.h>

// ---------------- problem constants (from reference) ----------------
#define NB   8           // graphs
#define NP   4096        // nodes per graph
#define NK   1024        // kept per graph
#define NTOT (NB*NP)     // 32768
#define NN   (NB*NK)     // 8192 pooled nodes
#define NE   262144      // edges
#define EPSV 1e-5f

typedef float v2f __attribute__((ext_vector_type(2)));
typedef float v8f __attribute__((ext_vector_type(8)));

// ---------------- init: remap=-1, degcnt=0, fill=0 ----------------
__global__ void k_init(int* remap, int* degcnt, int* fill) {
  int i = blockIdx.x * blockDim.x + threadIdx.x;
  if (i < NTOT) remap[i] = -1;
  if (i < NN) { degcnt[i] = 0; fill[i] = 0; }
}

// ---------------- scores: s = tanh(pos . p / ||p||) ----------------
__global__ void k_score(const float* __restrict__ pos, const float* __restrict__ p,
                        float* __restrict__ s) {
  int i = blockIdx.x * blockDim.x + threadIdx.x;
  if (i >= NTOT) return;
  float pn = sqrtf(p[0]*p[0] + p[1]*p[1] + p[2]*p[2]);
  float d = pos[i*3+0]*p[0] + pos[i*3+1]*p[1] + pos[i*3+2]*p[2];
  s[i] = tanhf(d / pn);
}

// ---------------- pad W1 [3,64] -> [4,64] with zero row ----------------
__global__ void k_padW1(const float* __restrict__ W1, float* __restrict__ W1p) {
  int t = threadIdx.x + blockIdx.x * blockDim.x;   // 256 threads total
  if (t >= 4*64) return;
  int k = t / 64, n = t - k*64;
  W1p[t] = (k < 3) ? W1[k*64 + n] : 0.0f;
}

// ordering: a before b  <=>  a.s > b.s, ties by lower index (matches lax.top_k)
__device__ __forceinline__ bool before_(float sa, int ia, float sb, int ib) {
  return (sa > sb) || (sa == sb && ia < ib);
}

// ---------------- per-graph top-k via LDS bitonic sort ----------------
// block = 1024 threads, grid = NB. Writes x0[NN,4] (4th column zero-padded).
__global__ void k_topk(const float* __restrict__ s, const float* __restrict__ pos,
                       float* __restrict__ x0, int* __restrict__ remap) {
  __shared__ float ss[NP];
  __shared__ int   si[NP];
  int g = blockIdx.x, t = threadIdx.x;
  for (int c = 0; c < 4; ++c) {
    int j = t + c*1024;
    int node = g*NP + j;
    ss[j] = s[node]; si[j] = node;
  }
  __syncthreads();
  for (int k = 2; k <= NP; k <<= 1) {
    for (int j = k >> 1; j > 0; j >>= 1) {
      for (int c = 0; c < 4; ++c) {
        int i = t + c*1024;
        int l = i ^ j;
        if (l > i) {
          bool up = ((i & k) == 0);
          bool sw = up ? before_(ss[l], si[l], ss[i], si[i])
                       : before_(ss[i], si[i], ss[l], si[l]);
          if (sw) {
            float tf = ss[i]; ss[i] = ss[l]; ss[l] = tf;
            int   ti = si[i]; si[i] = si[l]; si[l] = ti;
          }
        }
      }
      __syncthreads();
    }
  }
  // first NK entries are the kept nodes (descending score)
  int node = si[t];
  float v  = ss[t];
  int nid  = g*NK + t;
  x0[nid*4+0] = pos[node*3+0] * v;
  x0[nid*4+1] = pos[node*3+1] * v;
  x0[nid*4+2] = pos[node*3+2] * v;
  x0[nid*4+3] = 0.0f;                    // zero pad -> exact K=4 GEMM
  remap[node] = nid;
}

// ---------------- degree count over dst for valid edges ----------------
__global__ void k_deg(const int* __restrict__ ei, const int* __restrict__ remap,
                      int* __restrict__ degcnt) {
  int e = blockIdx.x * blockDim.x + threadIdx.x;
  if (e >= NE) return;
  int s = remap[ei[e]], d = remap[ei[NE + e]];
  if (s >= 0 && d >= 0) atomicAdd(&degcnt[d], 1);
}

// ---------------- exclusive scan of degcnt[NN] -> rowptr[NN+1] ----------------
__global__ void k_scan(const int* __restrict__ degcnt, int* __restrict__ rowptr) {
  __shared__ int part[1024];
  int t = threadIdx.x;
  int loc[8]; int sum = 0;
  for (int c = 0; c < 8; ++c) { loc[c] = sum; sum += degcnt[t*8 + c]; }
  part[t] = sum;
  __syncthreads();
  for (int off = 1; off < 1024; off <<= 1) {
    int v = (t >= off) ? part[t - off] : 0;
    __syncthreads();
    part[t] += v;
    __syncthreads();
  }
  int excl = (t == 0) ? 0 : part[t - 1];
  for (int c = 0; c < 8; ++c) rowptr[t*8 + c] = excl + loc[c];
  if (t == 1023) rowptr[NN] = part[1023];
}

// ---------------- dinv / self-loop norm ----------------
__global__ void k_dinv(const int* __restrict__ degcnt, float* __restrict__ dinv,
                       float* __restrict__ selfn) {
  int i = blockIdx.x * blockDim.x + threadIdx.x;
  if (i >= NN) return;
  float di = rsqrtf((float)degcnt[i] + 1.0f);
  dinv[i] = di;
  selfn[i] = di * di;
}

// ---------------- scatter valid edges into CSR slots ----------------
__global__ void k_scatter(const int* __restrict__ ei, const int* __restrict__ remap,
                          const int* __restrict__ rowptr, int* __restrict__ fill,
                          int* __restrict__ csrc, int* __restrict__ ceid) {
  int e = blockIdx.x * blockDim.x + threadIdx.x;
  if (e >= NE) return;
  int s = remap[ei[e]], d = remap[ei[NE + e]];
  if (s >= 0 && d >= 0) {
    int slot = rowptr[d] + atomicAdd(&fill[d], 1);
    csrc[slot] = s;
    ceid[slot] = e;
  }
}

// ---------------- per-node sort by edge id (determinism) + norms ----------------
__global__ void k_sortnorm(const int* __restrict__ rowptr, int* __restrict__ csrc,
                           int* __restrict__ ceid, const float* __restrict__ dinv,
                           float* __restrict__ cnorm) {
  int i = blockIdx.x * blockDim.x + threadIdx.x;
  if (i >= NN) return;
  int lo = rowptr[i], hi = rowptr[i+1];
  for (int a = lo + 1; a < hi; ++a) {          // insertion sort (degrees are tiny)
    int ke = ceid[a], ks = csrc[a];
    int b = a - 1;
    while (b >= lo && ceid[b] > ke) { ceid[b+1] = ceid[b]; csrc[b+1] = csrc[b]; --b; }
    ceid[b+1] = ke; csrc[b+1] = ks;
  }
  float di = dinv[i];
  for (int a = lo; a < hi; ++a) cnorm[a] = dinv[csrc[a]] * di;
}

// ---------------- fp32 WMMA GEMM: C[M,N] = A[M,K] @ B[K,N] ----------------
// Compile-time K,N (branch-free, fully resolvable addressing). One wave computes
// TM=4 vertical 16x16 tiles, reusing the B fragment across 4 WMMAs.
// Fragment layouts (ISA 7.12.2): A(16x4 f32) & B(4x16 f32): K = vgpr + 2*(lane>=16);
// C/D (16x16 f32): M = vgpr + 8*(lane>=16).
template<int K, int N, int TM>
__global__ void k_gemm_wmma(const float* __restrict__ A, const float* __restrict__ Bw,
                            float* __restrict__ C) {
  int wave = (blockIdx.x * blockDim.x + threadIdx.x) >> 5;
  int lane = threadIdx.x & 31;
  constexpr int tilesN = N / 16;
  int tm = (wave / tilesN) * TM;        // first M tile of this wave
  int tn = wave % tilesN;
  int h = lane >> 4, l = lane & 15;
  int col = tn*16 + l;                  // B/C column for this lane
  v8f acc[TM];
#pragma unroll
  for (int m = 0; m < TM; ++m) acc[m] = (v8f)0.0f;
  const float* __restrict__ arow[TM];
#pragma unroll
  for (int m = 0; m < TM; ++m) arow[m] = A + (size_t)((tm + m)*16 + l) * K;
#pragma unroll 4
  for (int k0 = 0; k0 < K; k0 += 4) {
    int ka = k0 + 2*h;
    v2f b;
    b.x = Bw[ka*N + col];
    b.y = Bw[(ka+1)*N + col];
#pragma unroll
    for (int m = 0; m < TM; ++m) {
      v2f a = *(const v2f*)(arow[m] + ka);   // 8B aligned: K%4==0, ka even
      acc[m] = __builtin_amdgcn_wmma_f32_16x16x4_f32(false, a, false, b,
                                                     (short)0, acc[m], false, false);
    }
  }
#pragma unroll
  for (int m = 0; m < TM; ++m)
#pragma unroll
    for (int v = 0; v < 8; ++v)
      C[(size_t)((tm + m)*16 + v + 8*h)*N + col] = acc[m][v];
}

// ---------------- aggregation: y = bias + selfn*h + sum_e norm*h[src] ----------------
__global__ void k_agg(const float* __restrict__ h, const int* __restrict__ rowptr,
                      const int* __restrict__ csrc, const float* __restrict__ cnorm,
                      const float* __restrict__ selfn, const float* __restrict__ bias,
                      float* __restrict__ y, int F) {
  int t = blockIdx.x * blockDim.x + threadIdx.x;
  if (t >= NN * F) return;
  int i = t / F, f = t - i*F;
  float acc = selfn[i] * h[i*F + f] + bias[f];
  int lo = rowptr[i], hi = rowptr[i+1];
  for (int e = lo; e < hi; ++e) acc += cnorm[e] * h[csrc[e]*F + f];
  y[t] = acc;
}

// ---------------- BN stats per feature (training-mode batch stats) ----------------
__global__ void k_bnstats(const float* __restrict__ y, const float* __restrict__ g,
                          const float* __restrict__ be, float* __restrict__ scale,
                          float* __restrict__ shift, int F) {
  __shared__ double rs[256], rs2[256];
  int f = blockIdx.x, t = threadIdx.x;
  double s = 0.0, s2 = 0.0;
  for (int i = t; i < NN; i += 256) {
    double v = (double)y[i*F + f];
    s += v; s2 += v*v;
  }
  rs[t] = s; rs2[t] = s2;
  __syncthreads();
  for (int o = 128; o > 0; o >>= 1) {
    if (t < o) { rs[t] += rs[t+o]; rs2[t] += rs2[t+o]; }
    __syncthreads();
  }
  if (t == 0) {
    double m   = rs[0]  / (double)NN;
    double var = rs2[0] / (double)NN - m*m;
    float rstd = rsqrtf((float)var + EPSV);
    float sc = g[f] * rstd;
    scale[f] = sc;
    shift[f] = be[f] - (float)m * sc;
  }
}

// ---------------- fused BN apply + ReLU ----------------
__global__ void k_bnrelu(const float* __restrict__ y, const float* __restrict__ scale,
                         const float* __restrict__ shift, float* __restrict__ x, int F) {
  int t = blockIdx.x * blockDim.x + threadIdx.x;
  if (t >= NN * F) return;
  int f = t % F;
  x[t] = fmaxf(0.0f, y[t] * scale[f] + shift[f]);
}

// ---------------- global sum pooling: [NN,512] -> [NB,512] ----------------
__global__ void k_pool(const float* __restrict__ x, float* __restrict__ xb) {
  int t = blockIdx.x * blockDim.x + threadIdx.x;   // t = b*512 + f
  if (t >= NB * 512) return;
  int b = t / 512, f = t - b*512;
  float acc = 0.0f;
  for (int j = 0; j < NK; ++j) acc += x[(b*NK + j)*512 + f];
  xb[t] = acc;
}

// ---------------- tiny dense layer (MLP head) ----------------
__global__ void k_dense(const float* __restrict__ in, const float* __restrict__ W,
                        const float* __restrict__ bias, float* __restrict__ out,
                        int Bn, int Fi, int Fo, int relu) {
  int t = blockIdx.x * blockDim.x + threadIdx.x;
  if (t >= Bn * Fo) return;
  int b = t / Fo, o = t - b*Fo;
  float acc = bias[o];
  for (int i = 0; i < Fi; ++i) acc += in[b*Fi + i] * W[i*Fo + o];
  out[t] = relu ? fmaxf(0.0f, acc) : acc;
}

// ---------------- log_softmax over rows of [NB,100] ----------------
__global__ void k_logsoftmax(const float* __restrict__ z, float* __restrict__ out) {
  __shared__ float red[128];
  int b = blockIdx.x, t = threadIdx.x;
  float m = -INFINITY;
  for (int c = t; c < 100; c += 128) m = fmaxf(m, z[b*100 + c]);
  red[t] = m; __syncthreads();
  for (int o = 64; o > 0; o >>= 1) { if (t < o) red[t] = fmaxf(red[t], red[t+o]); __syncthreads(); }
  m = red[0]; __syncthreads();
  float s = 0.0f;
  for (int c = t; c < 100; c += 128) s += expf(z[b*100 + c] - m);
  red[t] = s; __syncthreads();
  for (int o = 64; o > 0; o >>= 1) { if (t < o) red[t] += red[t+o]; __syncthreads(); }
  float lse = m + logf(red[0]);
  for (int c = t; c < 100; c += 128) out[b*100 + c] = z[b*100 + c] - lse;
}

// ---------------- host side ----------------
static inline char* ws_take(char*& p, size_t bytes) {
  char* r = p;
  p += (bytes + 255) & ~((size_t)255);
  return r;
}

extern "C" void kernel_launch(void* const* d_in, const int* in_sizes, int n_in,
                              void* d_out, int out_size, void* d_ws, size_t ws_size,
                              hipStream_t stream) {
  (void)in_sizes; (void)n_in; (void)out_size; (void)ws_size;
  const float* pos = (const float*)d_in[0];
  const int*   ei  = (const int*)d_in[1];
  // d_in[2] = batch (unused; layout is fixed)
  const float* p   = (const float*)d_in[3];
  const float* W[5]  = {(const float*)d_in[4],  (const float*)d_in[8],
                        (const float*)d_in[12], (const float*)d_in[16],
                        (const float*)d_in[20]};
  const float* bb[5] = {(const float*)d_in[5],  (const float*)d_in[9],
                        (const float*)d_in[13], (const float*)d_in[17],
                        (const float*)d_in[21]};
  const float* gg[5] = {(const float*)d_in[6],  (const float*)d_in[10],
                        (const float*)d_in[14], (const float*)d_in[18],
                        (const float*)d_in[22]};
  const float* bee[5]= {(const float*)d_in[7],  (const float*)d_in[11],
                        (const float*)d_in[15], (const float*)d_in[19],
                        (const float*)d_in[23]};
  const float* fw1 = (const float*)d_in[24]; const float* fb1 = (const float*)d_in[25];
  const float* fw2 = (const float*)d_in[26]; const float* fb2 = (const float*)d_in[27];
  const float* fw3 = (const float*)d_in[28]; const float* fb3 = (const float*)d_in[29];
  float* out = (float*)d_out;

  char* wp = (char*)d_ws;
  float* s      = (float*)ws_take(wp, (size_t)NTOT*4);
  int*   remap  = (int*)  ws_take(wp, (size_t)NTOT*4);
  float* bufA   = (float*)ws_take(wp, (size_t)NN*512*4);   // layer input / BN output
  float* bufH   = (float*)ws_take(wp, (size_t)NN*512*4);   // GEMM output
  float* bufY   = (float*)ws_take(wp, (size_t)NN*512*4);   // aggregated pre-BN
  int*   degcnt = (int*)  ws_take(wp, (size_t)NN*4);
  int*   fill   = (int*)  ws_take(wp, (size_t)NN*4);
  int*   rowptr = (int*)  ws_take(wp, (size_t)(NN+1)*4);
  float* dinv   = (float*)ws_take(wp, (size_t)NN*4);
  float* selfn  = (float*)ws_take(wp, (size_t)NN*4);
  int*   csrc   = (int*)  ws_take(wp, (size_t)NE*4);
  int*   ceid   = (int*)  ws_take(wp, (size_t)NE*4);
  float* cnorm  = (float*)ws_take(wp, (size_t)NE*4);
  float* scale  = (float*)ws_take(wp, 512*4);
  float* shift  = (float*)ws_take(wp, 512*4);
  float* W1p    = (float*)ws_take(wp, 4*64*4);             // zero-padded W1 [4,64]
  float* xb     = (float*)ws_take(wp, (size_t)NB*512*4);
  float* m1     = (float*)ws_take(wp, (size_t)NB*256*4);
  float* m2     = (float*)ws_take(wp, (size_t)NB*256*4);
  float* zlog   = (float*)ws_take(wp, (size_t)NB*100*4);

  // ---- graph construction (once) ----
  k_init  <<<NTOT/256, 256, 0, stream>>>(remap, degcnt, fill);
  k_score <<<NTOT/256, 256, 0, stream>>>(pos, p, s);
  k_padW1 <<<1, 256, 0, stream>>>(W[0], W1p);
  k_topk  <<<NB, 1024, 0, stream>>>(s, pos, bufA, remap);
  k_deg   <<<NE/256, 256, 0, stream>>>(ei, remap, degcnt);
  k_scan  <<<1, 1024, 0, stream>>>(degcnt, rowptr);
  k_dinv  <<<NN/256, 256, 0, stream>>>(degcnt, dinv, selfn);
  k_scatter<<<NE/256, 256, 0, stream>>>(ei, remap, rowptr, fill, csrc, ceid);
  k_sortnorm<<<NN/256, 256, 0, stream>>>(rowptr, csrc, ceid, dinv, cnorm);

  // ---- 5 x (GCNConv -> BN -> ReLU); GEMM grids: waves = (M/64)*(N/16), 8 waves/block
  const int Fout[5] = {64, 128, 128, 256, 512};
  for (int L = 0; L < 5; ++L) {
    int fo = Fout[L];
    int waves = (NN/64) * (fo/16);
    switch (L) {
      case 0: k_gemm_wmma<  4,  64, 4><<<waves/8, 256, 0, stream>>>(bufA, W1p, bufH); break;
      case 1: k_gemm_wmma< 64, 128, 4><<<waves/8, 256, 0, stream>>>(bufA, W[1], bufH); break;
      case 2: k_gemm_wmma<128, 128, 4><<<waves/8, 256, 0, stream>>>(bufA, W[2], bufH); break;
      case 3: k_gemm_wmma<128, 256, 4><<<waves/8, 256, 0, stream>>>(bufA, W[3], bufH); break;
      case 4: k_gemm_wmma<256, 512, 4><<<waves/8, 256, 0, stream>>>(bufA, W[4], bufH); break;
    }
    k_agg     <<<(NN*fo + 255)/256, 256, 0, stream>>>(bufH, rowptr, csrc, cnorm,
                                                      selfn, bb[L], bufY, fo);
    k_bnstats <<<fo, 256, 0, stream>>>(bufY, gg[L], bee[L], scale, shift, fo);
    k_bnrelu  <<<(NN*fo + 255)/256, 256, 0, stream>>>(bufY, scale, shift, bufA, fo);
  }

  // ---- pooling + MLP head + log_softmax ----
  k_pool <<<(NB*512 + 255)/256, 256, 0, stream>>>(bufA, xb);
  k_dense<<<(NB*256 + 255)/256, 256, 0, stream>>>(xb, fw1, fb1, m1, NB, 512, 256, 1);
  k_dense<<<(NB*256 + 255)/256, 256, 0, stream>>>(m1, fw2, fb2, m2, NB, 256, 256, 1);
  k_dense<<<(NB*100 + 255)/256, 256, 0, stream>>>(m2, fw3, fb3, zlog, NB, 256, 100, 0);
  k_logsoftmax<<<NB, 128, 0, stream>>>(zlog, out);
}